// AffinityLoss_32177894982349
// MI455X (gfx1250) — compile-verified
//
#include <hip/hip_runtime.h>

typedef __attribute__((ext_vector_type(2))) float v2f;
typedef __attribute__((ext_vector_type(8))) float v8f;

#define NCOLS      128
#define LDS_STRIDE 132          // 132 % 64 == 4 -> conflict-free b64 fragment reads
#define CHUNKS     32           // workgroups per batch
#define NB         16
#define DX         20
#define SY         4
#define WS_FLOATS  (NB * 3 * 256)

__global__ __launch_bounds__(256) void zero_ws_kernel(float* ws, int n) {
    int i = blockIdx.x * 256 + threadIdx.x;
    if (i < n) ws[i] = 0.0f;
}

__global__ __launch_bounds__(256) void gram_wmma_kernel(
    const float* __restrict__ x,   // (16, 20, N)
    const float* __restrict__ y,   // (16, 4, N)
    float* __restrict__ ws,        // (16, 3, 16, 16)
    int N, int nsteps)
{
    const int b    = blockIdx.x;
    const int tid  = threadIdx.x;
    const int lane = tid & 31;
    const int wave = tid >> 5;

    __shared__ float tile[32][LDS_STRIDE];

    // Zero pad rows 24..31 once (never written again).
    for (int idx = tid; idx < 8 * LDS_STRIDE; idx += 256)
        tile[24 + idx / LDS_STRIDE][idx % LDS_STRIDE] = 0.0f;

    const float* xb = x + (size_t)b * DX * N;
    const float* yb = y + (size_t)b * SY * N;

    v8f c00 = {}; v8f c01 = {}; v8f c11 = {};

    const int r    = lane & 15;         // M within 16-row block
    const int koff = (lane >> 4) * 2;   // K pair select per half-wave

    for (int step = blockIdx.y; step < nsteps; step += gridDim.y) {
        const int col0 = step * NCOLS;
        __syncthreads();   // previous panel fully consumed (and pad-zero visible)

        // Stage 24 x 128 panel: 768 float4 loads, 3 per thread, coalesced in-row.
        #pragma unroll
        for (int q = 0; q < 3; ++q) {
            const int idx = tid + 256 * q;
            const int row = idx >> 5;        // 0..23
            const int c4  = idx & 31;        // 0..31 (float4 index in row)
            const int col = col0 + 4 * c4;
            float4 v = make_float4(0.f, 0.f, 0.f, 0.f);
            if (col < N) {
                const float* src = (row < DX) ? (xb + (size_t)row * N + col)
                                              : (yb + (size_t)(row - DX) * N + col);
                v = *(const float4*)src;
            }
            float* dst = &tile[row][4 * c4];
            dst[0] = v.x; dst[1] = v.y; dst[2] = v.z; dst[3] = v.w;
        }
        __syncthreads();

        // Each wave consumes its private 16 columns: 4 WMMA k-steps of K=4.
        const int cbase = wave * 16;
        #pragma unroll
        for (int s = 0; s < 4; ++s) {
            const int k0 = cbase + 4 * s + koff;
            v2f f0 = *(const v2f*)&tile[r][k0];        // rows 0..15 fragment
            v2f f1 = *(const v2f*)&tile[16 + r][k0];   // rows 16..31 fragment
            // Same register image serves as A (rows) and B (rows^T): Gram tiles.
            c00 = __builtin_amdgcn_wmma_f32_16x16x4_f32(false, f0, false, f0,
                                                        (short)0, c00, false, false);
            c01 = __builtin_amdgcn_wmma_f32_16x16x4_f32(false, f0, false, f1,
                                                        (short)0, c01, false, false);
            c11 = __builtin_amdgcn_wmma_f32_16x16x4_f32(false, f1, false, f1,
                                                        (short)0, c11, false, false);
        }
    }

    // C/D layout: VGPR v, lanes 0-15 -> M=v, lanes 16-31 -> M=v+8; N = lane%16.
    const int mbase = (lane >> 4) * 8;
    const int n     = lane & 15;
    float* wsb = ws + (size_t)b * 768;
    #pragma unroll
    for (int v = 0; v < 8; ++v) {
        unsafeAtomicAdd(&wsb[0 * 256 + (mbase + v) * 16 + n], c00[v]);
        unsafeAtomicAdd(&wsb[1 * 256 + (mbase + v) * 16 + n], c01[v]);
        unsafeAtomicAdd(&wsb[2 * 256 + (mbase + v) * 16 + n], c11[v]);
    }
}

__global__ __launch_bounds__(256) void finalize_kernel(const float* __restrict__ ws,
                                                       float* __restrict__ out)
{
    float acc = 0.0f;
    for (int idx = threadIdx.x; idx < WS_FLOATS; idx += 256) {
        const int rem = idx % 768;
        const int T   = rem / 256;
        const int e   = rem % 256;
        const int m   = e / 16;
        const int n   = e % 16;
        float w;
        if (T == 0)        w = 1.0f;                              // XX block
        else if (T == 1)   w = (n < 4) ? 2.0f : -2.0f;            // (0,1) + its transpose
        else               w = ((m < 4) == (n < 4)) ? 1.0f : -1.0f; // (1,1) mixed block
        const float g = ws[idx];
        acc += w * g * g;
    }
    __shared__ float red[256];
    red[threadIdx.x] = acc;
    __syncthreads();
    for (int s = 128; s > 0; s >>= 1) {
        if (threadIdx.x < s) red[threadIdx.x] += red[threadIdx.x + s];
        __syncthreads();
    }
    if (threadIdx.x == 0) out[0] = red[0] / (float)NB;
}

extern "C" void kernel_launch(void* const* d_in, const int* in_sizes, int n_in,
                              void* d_out, int out_size, void* d_ws, size_t ws_size,
                              hipStream_t stream) {
    const float* x = (const float*)d_in[0];   // (16, 20, 257, 400) fp32
    const float* y = (const float*)d_in[1];   // (16,  4, 257, 400) fp32
    float* ws  = (float*)d_ws;
    float* out = (float*)d_out;

    const int N      = 257 * 400;             // 102800
    const int nsteps = (N + NCOLS - 1) / NCOLS;

    zero_ws_kernel<<<(WS_FLOATS + 255) / 256, 256, 0, stream>>>(ws, WS_FLOATS);
    gram_wmma_kernel<<<dim3(NB, CHUNKS), 256, 0, stream>>>(x, y, ws, N, nsteps);
    finalize_kernel<<<1, 256, 0, stream>>>(ws, out);
}